// DSTRformer_16054587752791
// MI455X (gfx1250) — compile-verified
//
#include <hip/hip_runtime.h>
#include <math.h>

// Problem constants (match reference)
constexpr int Bc = 8, Tc = 12, Nc = 512, Dc = 64, Hc = 4, HDc = 16, Lc = 4;
constexpr float THRESH = 0.1f;
constexpr float NEGV = -1000000000.0f;

typedef __attribute__((ext_vector_type(2))) float v2f;
typedef __attribute__((ext_vector_type(8))) float v8f;

// ---------------------------------------------------------------------------
// Kernel 1: lag MLP encoder, mean over lags, then a_src / a_tgt projections.
// block = 256 threads = 4 sub-blocks of 64; each sub-block owns one (b,n).
// a_tgt is stored transposed [b, d, n] so the NxN kernel reads it coalesced.
// ---------------------------------------------------------------------------
__global__ void lag_encode_kernel(const float* __restrict__ x,
                                  const float* __restrict__ lagW1,
                                  const float* __restrict__ lagb1,
                                  const float* __restrict__ lagW2,
                                  const float* __restrict__ lagb2,
                                  const float* __restrict__ sW1s,
                                  const float* __restrict__ sW1t,
                                  float* __restrict__ a_src,
                                  float* __restrict__ a_tgtT) {
  __shared__ float xrow[4][Dc];
  __shared__ float hbuf[4][Dc / 2];
  __shared__ float aggrow[4][Dc];
  const int sub = threadIdx.x >> 6;
  const int tid = threadIdx.x & 63;
  const int n = blockIdx.x * 4 + sub;
  const int b = blockIdx.y;

  float enc = 0.0f;
  for (int l = 0; l < Lc; ++l) {
    int tl = Tc - 1 - l; if (tl < 0) tl = 0;
    xrow[sub][tid] = x[(((size_t)b * Tc + tl) * Nc + n) * Dc + tid];
    __syncthreads();
    if (tid < Dc / 2) {
      const float* w1 = &lagW1[((size_t)l * (Dc / 2) + tid) * Dc];
      float acc = lagb1[l * (Dc / 2) + tid];
      #pragma unroll 8
      for (int d = 0; d < Dc; ++d) acc += xrow[sub][d] * w1[d];
      hbuf[sub][tid] = fmaxf(acc, 0.0f);
    }
    __syncthreads();
    const float* w2 = &lagW2[((size_t)l * Dc + tid) * (Dc / 2)];
    float acc2 = lagb2[l * Dc + tid];
    #pragma unroll 8
    for (int j = 0; j < Dc / 2; ++j) acc2 += hbuf[sub][j] * w2[j];
    enc += acc2;
    __syncthreads();
  }
  aggrow[sub][tid] = enc * (1.0f / Lc);
  __syncthreads();

  float as = 0.0f, at = 0.0f;
  const float* wsrow = &sW1s[tid * Dc];
  const float* wtrow = &sW1t[tid * Dc];
  #pragma unroll 8
  for (int d = 0; d < Dc; ++d) {
    float a = aggrow[sub][d];
    as += a * wsrow[d];
    at += a * wtrow[d];
  }
  a_src[((size_t)b * Nc + n) * Dc + tid] = as;
  a_tgtT[((size_t)b * Dc + tid) * Nc + n] = at;
}

// ---------------------------------------------------------------------------
// Kernel 2: adj[b,i,j] = thresholded sigmoid of edge strength, zero diagonal.
// grid = (N/256, N, B); a_src row cached in LDS; a_tgtT read coalesced.
// ---------------------------------------------------------------------------
__global__ void adj_kernel(const float* __restrict__ a_src,
                           const float* __restrict__ a_tgtT,
                           const float* __restrict__ sb1,
                           const float* __restrict__ sW2,
                           const float* __restrict__ sb2,
                           float* __restrict__ adj) {
  __shared__ float sA[Dc];
  __shared__ float sw[Dc];
  const int tid = threadIdx.x;
  const int j = blockIdx.x * 256 + tid;
  const int i = blockIdx.y;
  const int b = blockIdx.z;
  if (tid < Dc) {
    sA[tid] = a_src[((size_t)b * Nc + i) * Dc + tid] + sb1[tid];
    sw[tid] = sW2[tid];
  }
  __syncthreads();
  float acc = sb2[0];
  #pragma unroll 8
  for (int d = 0; d < Dc; ++d) {
    float t = sA[d] + a_tgtT[((size_t)b * Dc + d) * Nc + j];
    acc += fmaxf(t, 0.0f) * sw[d];
  }
  float s = 1.0f / (1.0f + expf(-acc));
  float v = (s > THRESH && i != j) ? s : 0.0f;
  adj[((size_t)b * Nc + i) * Nc + j] = v;
}

// ---------------------------------------------------------------------------
// Kernel 3: 512x512x512 f32 GEMM per batch via V_WMMA_F32_16X16X4_F32.
// Each wave32 computes a 16x64 strip (4 accumulators): the A fragment
// (one b64 per lane per K-step) is reused across 4 WMMAs, cutting VMEM
// issue per wmma ~1.6x vs. a 1-tile wave. Block = 128 threads = 4 full waves
// (EXEC all-ones around WMMA, no divergence).
// Lane layout per ISA: lanes 0-15 hold K={k0,k0+1}, lanes 16-31 K={k0+2,k0+3};
// D rows r (lanes 0-15) / r+8 (lanes 16-31).
// mode 0: out = A*B          (adj2 = adj*adj)
// mode 1: out = C0 + 0.5*C1 + 0.25*(A*B)   (enh = adj + .5 adj2 + .25 adj3)
// ---------------------------------------------------------------------------
__global__ void wmma_gemm_kernel(const float* __restrict__ A,
                                 const float* __restrict__ Bm,
                                 const float* __restrict__ C0,
                                 const float* __restrict__ C1,
                                 float* __restrict__ out, int mode) {
  const int lane = threadIdx.x & 31;
  const int wid  = threadIdx.x >> 5;            // 0..3
  const int lin  = blockIdx.x * 4 + wid;        // wave job 0..255
  const int ti   = lin >> 3;                    // tile row    (0..31)
  const int tj4  = lin & 7;                     // 4-tile col group (0..7)
  const int b    = blockIdx.y;

  const float* Ab = A  + (size_t)b * Nc * Nc;
  const float* Bb = Bm + (size_t)b * Nc * Nc;

  const int m  = lane & 15;                     // M for A / N for B / N for D
  const int kk = (lane >> 4) << 1;              // 0 or 2

  const int arow = ti * 16 + m;
  const int c0   = tj4 * 64 + m;                // base column of the strip

  v8f acc0 = {0.f,0.f,0.f,0.f,0.f,0.f,0.f,0.f};
  v8f acc1 = acc0, acc2 = acc0, acc3 = acc0;

  for (int k0 = 0; k0 < Nc; k0 += 4) {
    v2f av;
    av.x = Ab[(size_t)arow * Nc + k0 + kk];
    av.y = Ab[(size_t)arow * Nc + k0 + kk + 1];

    const float* br0 = &Bb[(size_t)(k0 + kk)     * Nc];
    const float* br1 = &Bb[(size_t)(k0 + kk + 1) * Nc];
    v2f bv0 = {br0[c0],      br1[c0]};
    v2f bv1 = {br0[c0 + 16], br1[c0 + 16]};
    v2f bv2 = {br0[c0 + 32], br1[c0 + 32]};
    v2f bv3 = {br0[c0 + 48], br1[c0 + 48]};

    // cover L2 latency of the strided B stream (lowers to global_prefetch_b8)
    if (k0 + 32 < Nc) {
      __builtin_prefetch(&Bb[(size_t)(k0 + 32 + kk) * Nc + c0], 0, 3);
      __builtin_prefetch(&Ab[(size_t)arow * Nc + k0 + 32 + kk], 0, 3);
    }

    acc0 = __builtin_amdgcn_wmma_f32_16x16x4_f32(false, av, false, bv0,
                                                 (short)0, acc0, false, false);
    acc1 = __builtin_amdgcn_wmma_f32_16x16x4_f32(false, av, false, bv1,
                                                 (short)0, acc1, false, false);
    acc2 = __builtin_amdgcn_wmma_f32_16x16x4_f32(false, av, false, bv2,
                                                 (short)0, acc2, false, false);
    acc3 = __builtin_amdgcn_wmma_f32_16x16x4_f32(false, av, false, bv3,
                                                 (short)0, acc3, false, false);
  }

  const int rbase = ti * 16 + ((lane < 16) ? 0 : 8);
  #pragma unroll
  for (int r = 0; r < 8; ++r) {
    const size_t rowoff = ((size_t)b * Nc + (rbase + r)) * Nc;
    float p[4] = {acc0[r], acc1[r], acc2[r], acc3[r]};
    #pragma unroll
    for (int t = 0; t < 4; ++t) {
      size_t idx = rowoff + c0 + t * 16;
      out[idx] = (mode == 0) ? p[t]
                             : (C0[idx] + 0.5f * C1[idx] + 0.25f * p[t]);
    }
  }
}

// ---------------------------------------------------------------------------
// Kernel 4: per-batch max of enh (all values >= 0).
// ---------------------------------------------------------------------------
__global__ void max_kernel(const float* __restrict__ enh, float* __restrict__ mx) {
  __shared__ float red[256];
  const int b = blockIdx.x;
  float lm = 0.0f;
  for (int idx = threadIdx.x; idx < Nc * Nc; idx += 256)
    lm = fmaxf(lm, enh[(size_t)b * Nc * Nc + idx]);
  red[threadIdx.x] = lm;
  __syncthreads();
  for (int s = 128; s > 0; s >>= 1) {
    if (threadIdx.x < s) red[threadIdx.x] = fmaxf(red[threadIdx.x], red[threadIdx.x + s]);
    __syncthreads();
  }
  if (threadIdx.x == 0) mx[b] = red[0];
}

// ---------------------------------------------------------------------------
// Kernel 5: QKV projections from x[:, T-1].  block = 64 per (n,b).
// ---------------------------------------------------------------------------
__global__ void qkv_kernel(const float* __restrict__ x,
                           const float* __restrict__ Wq, const float* __restrict__ bq,
                           const float* __restrict__ Wk, const float* __restrict__ bk,
                           const float* __restrict__ Wv, const float* __restrict__ bv,
                           float* __restrict__ qb, float* __restrict__ kb,
                           float* __restrict__ vb) {
  __shared__ float xr[Dc];
  const int n = blockIdx.x, b = blockIdx.y, d = threadIdx.x;
  xr[d] = x[(((size_t)b * Tc + (Tc - 1)) * Nc + n) * Dc + d];
  __syncthreads();
  float aq = bq[d], ak = bk[d], av = bv[d];
  #pragma unroll 8
  for (int k = 0; k < Dc; ++k) {
    float xv = xr[k];
    aq += xv * Wq[d * Dc + k];
    ak += xv * Wk[d * Dc + k];
    av += xv * Wv[d * Dc + k];
  }
  size_t o = ((size_t)b * Nc + n) * Dc + d;
  qb[o] = aq; kb[o] = ak; vb[o] = av;
}

// ---------------------------------------------------------------------------
// Kernel 6: masked softmax attention, one block (256 thr) per (i, h, b).
// Replicates reference: scores masked with NEG where enh==0, softmax,
// multiply by normalized enh, renormalize, weighted sum of V.
// ---------------------------------------------------------------------------
__global__ void attn_kernel(const float* __restrict__ qb,
                            const float* __restrict__ kb,
                            const float* __restrict__ vb,
                            const float* __restrict__ enh,
                            const float* __restrict__ mx,
                            float* __restrict__ ctx) {
  __shared__ float qrow[HDc];
  __shared__ float sc[Nc];
  __shared__ float red[256];
  const int i = blockIdx.x, h = blockIdx.y, b = blockIdx.z;
  const int t = threadIdx.x;
  if (t < HDc) qrow[t] = qb[((size_t)b * Nc + i) * Dc + h * HDc + t];
  __syncthreads();

  const float scale = 0.25f;   // 1/sqrt(HD=16)
  float lmax = NEGV;
  for (int j = t; j < Nc; j += 256) {
    float er = enh[((size_t)b * Nc + i) * Nc + j];
    float s;
    if (er == 0.0f) {
      s = NEGV;
    } else {
      s = 0.0f;
      const float* kr = &kb[((size_t)b * Nc + j) * Dc + h * HDc];
      #pragma unroll
      for (int d = 0; d < HDc; ++d) s += qrow[d] * kr[d];
      s *= scale;
    }
    sc[j] = s;
    lmax = fmaxf(lmax, s);
  }
  red[t] = lmax; __syncthreads();
  for (int s2 = 128; s2 > 0; s2 >>= 1) {
    if (t < s2) red[t] = fmaxf(red[t], red[t + s2]);
    __syncthreads();
  }
  const float gmax = red[0]; __syncthreads();

  float lsum = 0.0f;
  for (int j = t; j < Nc; j += 256) {
    float e = expf(sc[j] - gmax);
    sc[j] = e;
    lsum += e;
  }
  red[t] = lsum; __syncthreads();
  for (int s2 = 128; s2 > 0; s2 >>= 1) {
    if (t < s2) red[t] += red[t + s2];
    __syncthreads();
  }
  const float denom = red[0]; __syncthreads();

  const float invMx = 1.0f / (mx[b] + 1e-8f);
  float lw = 0.0f;
  for (int j = t; j < Nc; j += 256) {
    float er = enh[((size_t)b * Nc + i) * Nc + j];
    float w = (sc[j] / denom) * er * invMx;
    sc[j] = w;
    lw += w;
  }
  red[t] = lw; __syncthreads();
  for (int s2 = 128; s2 > 0; s2 >>= 1) {
    if (t < s2) red[t] += red[t + s2];
    __syncthreads();
  }
  const float invW = 1.0f / (red[0] + 1e-8f); __syncthreads();

  // ctx: 16 output dims x 16 j-segments across 256 threads
  const int d = t & 15, seg = t >> 4;
  float acc = 0.0f;
  for (int j = seg * 32; j < seg * 32 + 32; ++j)
    acc += sc[j] * vb[((size_t)b * Nc + j) * Dc + h * HDc + d];
  red[t] = acc; __syncthreads();
  if (t < HDc) {
    float s = 0.0f;
    #pragma unroll
    for (int g = 0; g < 16; ++g) s += red[g * 16 + t];
    ctx[((size_t)b * Nc + i) * Dc + h * HDc + t] = s * invW;
  }
}

// ---------------------------------------------------------------------------
// Kernel 7: output projection out = ctx @ Wo^T + bo.  block = 64 per (n,b).
// ---------------------------------------------------------------------------
__global__ void outproj_kernel(const float* __restrict__ ctx,
                               const float* __restrict__ Wo,
                               const float* __restrict__ bo,
                               float* __restrict__ out) {
  __shared__ float cr[Dc];
  const int n = blockIdx.x, b = blockIdx.y, d = threadIdx.x;
  cr[d] = ctx[((size_t)b * Nc + n) * Dc + d];
  __syncthreads();
  float acc = bo[d];
  #pragma unroll 8
  for (int k = 0; k < Dc; ++k) acc += cr[k] * Wo[d * Dc + k];
  out[((size_t)b * Nc + n) * Dc + d] = acc;
}

// ---------------------------------------------------------------------------
extern "C" void kernel_launch(void* const* d_in, const int* in_sizes, int n_in,
                              void* d_out, int out_size, void* d_ws, size_t ws_size,
                              hipStream_t stream) {
  const float* x     = (const float*)d_in[0];
  const float* lagW1 = (const float*)d_in[1];
  const float* lagb1 = (const float*)d_in[2];
  const float* lagW2 = (const float*)d_in[3];
  const float* lagb2 = (const float*)d_in[4];
  const float* sW1s  = (const float*)d_in[5];
  const float* sW1t  = (const float*)d_in[6];
  const float* sb1   = (const float*)d_in[7];
  const float* sW2   = (const float*)d_in[8];
  const float* sb2   = (const float*)d_in[9];
  const float* Wq    = (const float*)d_in[10];
  const float* bq    = (const float*)d_in[11];
  const float* Wk    = (const float*)d_in[12];
  const float* bk    = (const float*)d_in[13];
  const float* Wv    = (const float*)d_in[14];
  const float* bv    = (const float*)d_in[15];
  const float* Wo    = (const float*)d_in[16];
  const float* bo    = (const float*)d_in[17];
  float* out = (float*)d_out;

  // workspace layout (floats)
  float* w = (float*)d_ws;
  const size_t szBND = (size_t)Bc * Nc * Dc;   // 262144
  const size_t szBNN = (size_t)Bc * Nc * Nc;   // 2097152
  float* a_src  = w;                 w += szBND;
  float* a_tgtT = w;                 w += szBND;   // [b, d, n]
  float* adj    = w;                 w += szBNN;
  float* adj2   = w;                 w += szBNN;
  float* enh    = w;                 w += szBNN;
  float* mxb    = w;                 w += 64;
  float* qb     = w;                 w += szBND;
  float* kb     = w;                 w += szBND;
  float* vb     = w;                 w += szBND;
  float* ctx    = w;                 w += szBND;

  // 1. lag encoder + aggregate + src/tgt projections
  lag_encode_kernel<<<dim3(Nc / 4, Bc), 256, 0, stream>>>(
      x, lagW1, lagb1, lagW2, lagb2, sW1s, sW1t, a_src, a_tgtT);

  // 2. adjacency from pairwise strengths
  adj_kernel<<<dim3(Nc / 256, Nc, Bc), 256, 0, stream>>>(
      a_src, a_tgtT, sb1, sW2, sb2, adj);

  // 3. adj2 = adj @ adj   (WMMA f32, 16x64 strip per wave)
  wmma_gemm_kernel<<<dim3(64, Bc), 128, 0, stream>>>(
      adj, adj, nullptr, nullptr, adj2, 0);

  // 4. enh = adj + 0.5*adj2 + 0.25*(adj2 @ adj)   (WMMA f32)
  wmma_gemm_kernel<<<dim3(64, Bc), 128, 0, stream>>>(
      adj2, adj, adj, adj2, enh, 1);

  // 5. per-batch max of enh
  max_kernel<<<dim3(Bc), 256, 0, stream>>>(enh, mxb);

  // 6. QKV projections from last timestep
  qkv_kernel<<<dim3(Nc, Bc), Dc, 0, stream>>>(x, Wq, bq, Wk, bk, Wv, bv, qb, kb, vb);

  // 7. masked attention
  attn_kernel<<<dim3(Nc, Hc, Bc), 256, 0, stream>>>(qb, kb, vb, enh, mxb, ctx);

  // 8. output projection
  outproj_kernel<<<dim3(Nc, Bc), Dc, 0, stream>>>(ctx, Wo, bo, out);
}